// Playlist_Model_53833120088400
// MI455X (gfx1250) — compile-verified
//
#include <hip/hip_runtime.h>
#include <hip/hip_bf16.h>

// ---------------------------------------------------------------------------
// Playlist two-tower query model, CDNA5 (gfx1250), wave32.
//   Phase 1: embedding gathers + mean pool  -> x0 [2048,480]   (bandwidth bound,
//            ~886 MB of random 128B-row gathers => ~40us floor @ 23.3 TB/s)
//   Phase 2: low-rank DCN cross (rank 5), in place on x0
//   Phase 3: MLP 480->512->256->128 via V_WMMA_F32_16X16X4_F32 (f32-exact,
//            4 M-tiles per wave sharing the B fragment)
//   Phase 4: row-wise L2 normalize -> d_out [2048,128]
// ---------------------------------------------------------------------------

#define BATCH 2048
#define SEQL  375
#define EMB   32
#define DIN   480
#define H1    512
#define H2    256
#define H3    128
#define PROJ  5
#define MT    4          // M-tiles (of 16 rows) per wave in the GEMM

typedef __attribute__((ext_vector_type(2))) float v2f;
typedef __attribute__((ext_vector_type(4))) float v4f;
typedef __attribute__((ext_vector_type(8))) float v8f;

// Global (addrspace 1) pointer types: force global_load_* (LOADcnt only),
// never flat_load_* (which couples LOADcnt+DScnt and uses the LDS path).
typedef const __attribute__((address_space(1))) float* gfp;
typedef const __attribute__((address_space(1))) int*   gip;
typedef const __attribute__((address_space(1))) v2f*   gv2p;
typedef const __attribute__((address_space(1))) v4f*   gv4p;

struct GatherPtrs {
    const int*   idx[15];
    const float* tab[15];
};

// ---------------------------------------------------------------------------
// Phase 1: one block per batch row; 16 waves; wave f handles feature f.
// Features 0-5: scalar gather [B].  Features 6-14: sequence gather [B,L] + mean.
// Lane layout for sequence features: group = lane>>3 handles positions
// l = group, group+4, ... ; sub = lane&7 covers 4 of the 32 emb floats (b128).
// ---------------------------------------------------------------------------
__global__ __launch_bounds__(512) void gather_pool_kernel(GatherPtrs p, float* __restrict__ x0)
{
    const int b    = blockIdx.x;
    const int wave = threadIdx.x >> 5;
    const int lane = threadIdx.x & 31;
    if (wave >= 15) return;

    float* dst = x0 + (size_t)b * DIN + wave * EMB;

    if (wave < 6) {
        // scalar feature: single row gather
        if (lane < 8) {
            const int ix = *(gip)(p.idx[wave] + b);
            v4f v = *(gv4p)(p.tab[wave] + (size_t)ix * EMB + lane * 4);
            *(v4f*)(dst + lane * 4) = v;
        }
    } else {
        const int*   idx = p.idx[wave] + (size_t)b * SEQL;
        const float* tab = p.tab[wave];
        const int group = lane >> 3;
        const int sub   = lane & 7;

        v4f acc = {0.f, 0.f, 0.f, 0.f};
        #pragma unroll 2
        for (int l = group; l < SEQL; l += 4) {
            // stream the index array ahead of use (global_prefetch_b8)
            __builtin_prefetch(idx + l + 32, 0, 1);
            const int ix = *(gip)(idx + l);
            v4f v = *(gv4p)(tab + (size_t)ix * EMB + sub * 4);
            acc.x += v.x; acc.y += v.y; acc.z += v.z; acc.w += v.w;
        }
        // reduce the 4 lane-groups (positions mod 4) down to one sum
        acc.x += __shfl_xor(acc.x, 8, 32);
        acc.y += __shfl_xor(acc.y, 8, 32);
        acc.z += __shfl_xor(acc.z, 8, 32);
        acc.w += __shfl_xor(acc.w, 8, 32);
        acc.x += __shfl_xor(acc.x, 16, 32);
        acc.y += __shfl_xor(acc.y, 16, 32);
        acc.z += __shfl_xor(acc.z, 16, 32);
        acc.w += __shfl_xor(acc.w, 16, 32);

        if (lane < 8) {
            const float inv = 1.0f / (float)SEQL;
            v4f r;
            r.x = acc.x * inv; r.y = acc.y * inv;
            r.z = acc.z * inv; r.w = acc.w * inv;
            *(v4f*)(dst + sub * 4) = r;
        }
    }
}

// ---------------------------------------------------------------------------
// Phase 2: DCN cross layer, in place:  x = x0 * ((x0 @ V) @ U + b) + x0
// One block (256 threads) per batch row.
// ---------------------------------------------------------------------------
__global__ __launch_bounds__(256) void cross_kernel(float* __restrict__ x0,
                                                    const float* __restrict__ V,
                                                    const float* __restrict__ U,
                                                    const float* __restrict__ bvec)
{
    const int b   = blockIdx.x;
    const int tid = threadIdx.x;

    __shared__ float sx[DIN];
    __shared__ float red[256];
    __shared__ float t[PROJ];

    float* row = x0 + (size_t)b * DIN;
    for (int i = tid; i < DIN; i += 256) sx[i] = row[i];
    __syncthreads();

    float p[PROJ] = {0.f, 0.f, 0.f, 0.f, 0.f};
    for (int k = tid; k < DIN; k += 256) {
        const float xv = sx[k];
        #pragma unroll
        for (int j = 0; j < PROJ; ++j) p[j] += xv * V[k * PROJ + j];
    }
    #pragma unroll
    for (int j = 0; j < PROJ; ++j) {
        red[tid] = p[j];
        __syncthreads();
        for (int s = 128; s > 0; s >>= 1) {
            if (tid < s) red[tid] += red[tid + s];
            __syncthreads();
        }
        if (tid == 0) t[j] = red[0];
        __syncthreads();
    }

    for (int c = tid; c < DIN; c += 256) {
        float proj = bvec[c];
        #pragma unroll
        for (int j = 0; j < PROJ; ++j) proj += t[j] * U[j * DIN + c];
        const float xv = sx[c];
        row[c] = xv * proj + xv;
    }
}

// ---------------------------------------------------------------------------
// Phase 3: C = act(A[M,K] @ B[K,N] + bias).  One wave computes MT (=4)
// vertically-stacked 16x16 output tiles (64 rows x 16 cols), sharing one
// B fragment per K-step across MT WMMAs: 4 x global_load_b64 (A) +
// 2 x global_load_b32 (B) per 4 x V_WMMA_F32_16X16X4_F32.  All fragment
// loads are issued before the WMMA burst so they form one vmem clause.
//
// A VGPR layout (16x4):  lanes 0-15 hold K=k,k+1 ; lanes 16-31 hold K=k+2,k+3.
// B VGPR layout (4x16):  mirrors A (VGPR0: K=k / k+2 ; VGPR1: K=k+1 / k+3).
// C layout: VGPR i -> M = i (+8 for upper half-wave), N = lane & 15.
// Launched with exactly (M/64)*(N/16) waves: EXEC all-1s, no divergence.
// ---------------------------------------------------------------------------
__global__ __launch_bounds__(256) void gemm_wmma_kernel(const float* __restrict__ A,
                                                        const float* __restrict__ B,
                                                        const float* __restrict__ bias,
                                                        float* __restrict__ C,
                                                        int M, int N, int K, int relu)
{
    const int wave   = (blockIdx.x * blockDim.x + threadIdx.x) >> 5;
    const int lane   = threadIdx.x & 31;
    const int tilesN = N >> 4;
    const int tmBase = (wave / tilesN) * (MT * 16);
    const int tn     = wave % tilesN;

    const int rsub  = lane & 15;
    const int col   = tn * 16 + rsub;
    const int khalf = (lane >> 4) << 1;        // 0 for lanes 0-15, 2 for 16-31

    const float* Ap[MT];
    #pragma unroll
    for (int i = 0; i < MT; ++i)
        Ap[i] = A + (size_t)(tmBase + i * 16 + rsub) * K + khalf;

    const float* Bp0 = B + (size_t)khalf * N + col;
    const float* Bp1 = Bp0 + N;
    const size_t bstep = (size_t)4 * N;

    v8f acc[MT] = {};
    for (int k = 0; k < K; k += 4) {
        // --- load clause: B fragment + all MT A fragments -------------------
        v2f bb;
        bb.x = *(gfp)Bp0;
        bb.y = *(gfp)Bp1;
        v2f a[MT];
        #pragma unroll
        for (int i = 0; i < MT; ++i)
            a[i] = *(gv2p)(Ap[i] + k);
        Bp0 += bstep;
        Bp1 += bstep;
        // --- WMMA burst -----------------------------------------------------
        #pragma unroll
        for (int i = 0; i < MT; ++i)
            acc[i] = __builtin_amdgcn_wmma_f32_16x16x4_f32(
                         /*neg_a=*/false, a[i], /*neg_b=*/false, bb,
                         /*c_mod=*/(short)0, acc[i],
                         /*reuse_a=*/false, /*reuse_b=*/false);
    }

    const float bv = *(gfp)(bias + col);
    #pragma unroll
    for (int i = 0; i < MT; ++i) {
        const int mbase = tmBase + i * 16 + ((lane >> 4) << 3);
        #pragma unroll
        for (int r = 0; r < 8; ++r) {
            float v = acc[i][r] + bv;
            if (relu) v = fmaxf(v, 0.f);
            C[(size_t)(mbase + r) * N + col] = v;
        }
    }
}

// ---------------------------------------------------------------------------
// Phase 4: row L2 normalize, block (128 threads) per row.
// ---------------------------------------------------------------------------
__global__ __launch_bounds__(128) void l2norm_kernel(const float* __restrict__ h,
                                                     float* __restrict__ out)
{
    const int b   = blockIdx.x;
    const int tid = threadIdx.x;
    const float v = *(gfp)(h + (size_t)b * H3 + tid);
    float ss = v * v;
    for (int m = 16; m > 0; m >>= 1) ss += __shfl_xor(ss, m, 32);

    __shared__ float wsum[4];
    if ((tid & 31) == 0) wsum[tid >> 5] = ss;
    __syncthreads();
    const float tot  = wsum[0] + wsum[1] + wsum[2] + wsum[3];
    const float norm = sqrtf(fmaxf(tot, 1e-12f));
    out[(size_t)b * H3 + tid] = v / norm;
}

// ---------------------------------------------------------------------------
// Launcher.  Input order (setup_inputs): 15 idx, 15 tables, cross_V, cross_U,
// cross_b, W1, b1, W2, b2, W3, b3.  Output: [2048,128] f32.
// ---------------------------------------------------------------------------
extern "C" void kernel_launch(void* const* d_in, const int* in_sizes, int n_in,
                              void* d_out, int out_size, void* d_ws, size_t ws_size,
                              hipStream_t stream)
{
    (void)in_sizes; (void)n_in; (void)out_size; (void)ws_size;

    GatherPtrs gp;
    for (int i = 0; i < 15; ++i) {
        gp.idx[i] = (const int*)  d_in[i];
        gp.tab[i] = (const float*)d_in[15 + i];
    }
    const float* cross_V = (const float*)d_in[30];
    const float* cross_U = (const float*)d_in[31];
    const float* cross_b = (const float*)d_in[32];
    const float* W1 = (const float*)d_in[33];
    const float* b1 = (const float*)d_in[34];
    const float* W2 = (const float*)d_in[35];
    const float* b2 = (const float*)d_in[36];
    const float* W3 = (const float*)d_in[37];
    const float* b3 = (const float*)d_in[38];

    // workspace layout (bytes); ~10.2 MB total, fully overwritten every call
    char*  ws = (char*)d_ws;
    float* x0 = (float*)(ws);                                   // 2048*480*4 = 3,932,160
    float* h1 = (float*)(ws + (size_t)BATCH * DIN * 4);         // 2048*512*4 = 4,194,304
    float* h2 = (float*)(ws + (size_t)BATCH * (DIN + H1) * 4);  // 2048*256*4 = 2,097,152

    // Phase 1: gathers + mean pool -> x0
    gather_pool_kernel<<<BATCH, 512, 0, stream>>>(gp, x0);

    // Phase 2: cross layer in place on x0
    cross_kernel<<<BATCH, 256, 0, stream>>>(x0, cross_V, cross_U, cross_b);

    // Phase 3: MLP tower.  Each wave owns a 64x16 output panel; 8 waves/block.
    {
        int waves = (BATCH / (16 * MT)) * (H1 / 16);   // 32*32 = 1024
        gemm_wmma_kernel<<<waves / 8, 256, 0, stream>>>(x0, W1, b1, h1, BATCH, H1, DIN, 1);
    }
    {
        int waves = (BATCH / (16 * MT)) * (H2 / 16);   // 32*16 = 512
        gemm_wmma_kernel<<<waves / 8, 256, 0, stream>>>(h1, W2, b2, h2, BATCH, H2, H1, 1);
    }
    // h3 reuses h1's space (free after gemm2)
    float* h3 = h1;
    {
        int waves = (BATCH / (16 * MT)) * (H3 / 16);   // 32*8 = 256
        gemm_wmma_kernel<<<waves / 8, 256, 0, stream>>>(h2, W3, b3, h3, BATCH, H3, H2, 0);
    }

    // Phase 4: L2 normalize -> d_out
    l2norm_kernel<<<BATCH, 128, 0, stream>>>(h3, (float*)d_out);
}